// Upsampler_19052474925532
// MI455X (gfx1250) — compile-verified
//
#include <hip/hip_runtime.h>
#include <stdint.h>

// Problem constants (from the reference)
#define HH 299
#define WW 299
#define S_IMGS 32
#define G_BOX 16
#define CCH 3
#define BORDER 3

// Tiling
#define ROWS 8            // output rows per block
#define TILE_MAXR 10      // upsampling => <= ROWS+2 input rows needed
#define TILE_ELEMS 3000   // >= TILE_MAXR * 298

__device__ __forceinline__ int imin(int a, int b) { return a < b ? a : b; }
__device__ __forceinline__ int imax(int a, int b) { return a > b ? a : b; }

// PyTorch/JAX bilinear (align_corners=False) source coords, fp32, matching reference:
//   src = (i + 0.5) * e / 299 - 0.5 ; clip to [0, e-1] ; i0 = floor ; i1 = min(i0+1, e-1)
__device__ __forceinline__ void axis_coord(int i, int lo, int e,
                                           int& g0, int& g1, float& fr) {
    float ef  = (float)e;
    float src = ((float)i + 0.5f) * ef / 299.0f - 0.5f;
    src = fminf(fmaxf(src, 0.0f), ef - 1.0f);
    int i0 = (int)floorf(src);
    i0 = imin(i0, e - 1);
    int i1 = imin(i0 + 1, e - 1);
    fr = src - (float)i0;
    g0 = lo + i0;
    g1 = lo + i1;
}

__global__ __launch_bounds__(256)
void Upsampler_crop_resize_kernel(const float* __restrict__ x,
                                  const int*   __restrict__ f,
                                  float*       __restrict__ out) {
    __shared__ int   sY0[WW];
    __shared__ int   sY1[WW];
    __shared__ float sFY[WW];
    __shared__ int   sX0[ROWS];
    __shared__ int   sX1[ROWS];
    __shared__ float sFX[ROWS];
    __shared__ float sTile[TILE_ELEMS];

    const int chunk = blockIdx.x;           // output-row chunk
    const int c     = blockIdx.y;           // channel
    const int sg    = blockIdx.z;           // image*16 + box
    const int si    = sg >> 4;
    const int tid   = threadIdx.x;
    const int nthr  = blockDim.x;

    // ---- box -> crop window (uniform scalars) ----
    const int* bp = f + (size_t)sg * 4;
    const int b0 = bp[0], b1 = bp[1], b2 = bp[2], b3 = bp[3];
    const int tlx = imax(b0 - BORDER, 0);
    const int tly = imax(b1 - BORDER, 0);
    const int brx = imin(b2 + BORDER, HH - 1);
    const int bry = imin(b3 + BORDER, WW - 1);
    const int hc  = imax(brx - tlx, 1);
    const int wc  = imax(bry - tly, 1);

    const int r0 = chunk * ROWS;
    const int rn = imin(ROWS, HH - r0);     // valid output rows in this chunk

    // ---- build coordinate tables in LDS ----
    for (int j = tid; j < WW; j += nthr) {
        int g0, g1; float fr;
        axis_coord(j, tly, wc, g0, g1, fr);
        sY0[j] = g0; sY1[j] = g1; sFY[j] = fr;
    }
    for (int j = tid; j < rn; j += nthr) {
        int g0, g1; float fr;
        axis_coord(r0 + j, tlx, hc, g0, g1, fr);
        sX0[j] = g0; sX1[j] = g1; sFX[j] = fr;
    }
    __syncthreads();

    // ---- staging window (monotone coords => contiguous ranges) ----
    const int rLo = sX0[0];
    const int rHi = sX1[rn - 1];
    const int cLo = sY0[0];
    const int cHi = sY1[WW - 1];
    const int nR  = imin(rHi - rLo + 1, TILE_MAXR);
    const int nC  = cHi - cLo + 1;          // <= 298

    // ---- stage input sub-tile into LDS via CDNA5 async global->LDS ----
    const float* srcBase = x + (((size_t)si * CCH + c) * HH + rLo) * WW + cLo;
    for (int ir = 0; ir < nR; ++ir) {
        const float* rowp = srcBase + (size_t)ir * WW;
        float*       ldsp = sTile + ir * nC;
        for (int ic = tid; ic < nC; ic += nthr) {
#if defined(__gfx1250__) && __has_builtin(__builtin_amdgcn_global_load_async_to_lds_b32)
            __builtin_amdgcn_global_load_async_to_lds_b32(
                (__attribute__((address_space(1))) int*)(rowp + ic),
                (__attribute__((address_space(3))) int*)(ldsp + ic),
                0, 0);
#else
            ldsp[ic] = rowp[ic];
#endif
        }
    }
#if defined(__gfx1250__) && __has_builtin(__builtin_amdgcn_s_wait_asynccnt)
    __builtin_amdgcn_s_wait_asynccnt(0);
#elif defined(__gfx1250__)
    asm volatile("s_wait_asynccnt 0" ::: "memory");
#endif
    __syncthreads();

    // ---- bilinear lerp from LDS; non-temporal (streaming) stores ----
    const int    nPix    = rn * WW;
    const size_t outBase = (((size_t)sg * CCH + c) * HH + r0) * WW;
    for (int t = tid; t < nPix; t += nthr) {
        const int j  = t / WW;              // const divisor -> mul/shift
        const int jc = t - j * WW;
        const int x0l = sX0[j] - rLo;
        const int x1l = sX1[j] - rLo;
        const float fx = sFX[j];
        const int y0l = sY0[jc] - cLo;
        const int y1l = sY1[jc] - cLo;
        const float fy = sFY[jc];

        const float v00 = sTile[x0l * nC + y0l];
        const float v01 = sTile[x0l * nC + y1l];
        const float v10 = sTile[x1l * nC + y0l];
        const float v11 = sTile[x1l * nC + y1l];

        const float top = v00 * (1.0f - fy) + v01 * fy;
        const float bot = v10 * (1.0f - fy) + v11 * fy;
        const float res = top * (1.0f - fx) + bot * fx;
        // Output is 549 MB write-once (>> 192 MB L2): stream it past the
        // cache so L2 keeps the (reused) input image and staged tiles.
        __builtin_nontemporal_store(res, &out[outBase + t]);
    }
}

extern "C" void kernel_launch(void* const* d_in, const int* in_sizes, int n_in,
                              void* d_out, int out_size, void* d_ws, size_t ws_size,
                              hipStream_t stream) {
    const float* x = (const float*)d_in[0];   // (32, 3, 299, 299) fp32
    const int*   f = (const int*)d_in[1];     // (32, 16, 4) int32
    float* out = (float*)d_out;               // (32, 16, 3, 299, 299) fp32

    dim3 grid((HH + ROWS - 1) / ROWS, CCH, S_IMGS * G_BOX);
    dim3 block(256);
    hipLaunchKernelGGL(Upsampler_crop_resize_kernel, grid, block, 0, stream, x, f, out);
}